// TXCDRTied_90984587198478
// MI455X (gfx1250) — compile-verified
//
#include <hip/hip_runtime.h>
#include <hip/hip_bf16.h>

// Problem constants
#define B_SZ   32
#define S_SZ   65536
#define T_SZ   4
#define D_IN   768
#define KDIM   3072          // T_SZ * D_IN, contracted dim of encoder GEMM
#define KTOP   64
#define KC     256           // k-chunk staged in LDS per outer iter
#define XLDSTR 260           // padded LDS row stride (floats): 4-float pad -> conflict-free ds_load_b64

typedef __attribute__((ext_vector_type(2))) float v2f;
typedef __attribute__((ext_vector_type(8))) float v8f;

// ---------------------------------------------------------------------------
// Kernel 1: encode GEMM  pre[b][s] = sum_k X[b,k] * W[s,k] + b_enc[s]
// fp32 WMMA 16x16x4.  A = W tile (16 s-rows x 4 k), B = X^T tile (4 k x 16 b).
// grid: 512 blocks x 256 threads (8 waves); wave -> 16 s-rows, both 16-wide b tiles.
// ---------------------------------------------------------------------------
__global__ __launch_bounds__(256) void sae_encode_kernel(
    const float* __restrict__ x, const float* __restrict__ W,
    const float* __restrict__ b_enc, float* __restrict__ pre)
{
    __shared__ float xs[B_SZ * XLDSTR];

    const int tid  = threadIdx.x;
    const int lane = tid & 31;
    const int wave = tid >> 5;                // 0..7
    const int s0   = blockIdx.x * 128 + wave * 16;
    const int m    = lane & 15;               // A row within tile / B col (batch)
    const int hi   = lane >> 4;               // 0 or 1
    const int kh   = hi * 2;                  // k pair held by this half-wave

    const float* wrow = W + (size_t)(s0 + m) * KDIM + kh;

    v8f c0 = {};   // batch cols 0..15
    v8f c1 = {};   // batch cols 16..31

    for (int kc = 0; kc < KDIM; kc += KC) {
        __syncthreads();
        // stage X[0..31][kc .. kc+KC) into LDS: 32*KC floats = 2048 float4
        #pragma unroll
        for (int q = 0; q < 8; ++q) {
            int i4  = q * 256 + tid;          // float4 index
            int bb  = i4 >> 6;                // KC/4 = 64 float4 per row
            int kk4 = i4 & 63;
            float4 v = *(const float4*)(x + (size_t)bb * KDIM + kc + kk4 * 4);
            *(float4*)(&xs[bb * XLDSTR + kk4 * 4]) = v;
        }
        __syncthreads();

        // prefetch next W chunk for this lane's row (speculative; OOB is dropped)
        __builtin_prefetch(wrow + kc + KC, 0, 1);

        #pragma unroll 4
        for (int k = 0; k < KC; k += 4) {
            v2f a  = *(const v2f*)(wrow + kc + k);                       // W[s0+m][kc+k+kh .. +1]
            v2f b0 = *(const v2f*)(&xs[m        * XLDSTR + k + kh]);     // X[m   ][kc+k+kh .. +1]
            v2f b1 = *(const v2f*)(&xs[(m + 16) * XLDSTR + k + kh]);     // X[m+16][...]
            c0 = __builtin_amdgcn_wmma_f32_16x16x4_f32(false, a, false, b0,
                                                       (short)0, c0, false, false);
            c1 = __builtin_amdgcn_wmma_f32_16x16x4_f32(false, a, false, b1,
                                                       (short)0, c1, false, false);
        }
    }

    // D layout: element v of accumulator <-> (s = s0 + v + 8*hi, b = lane%16 [+16 for c1])
    #pragma unroll
    for (int v = 0; v < 8; ++v) {
        int srow = s0 + v + 8 * hi;
        float be = b_enc[srow];
        pre[(size_t)m        * S_SZ + srow] = c0[v] + be;
        pre[(size_t)(m + 16) * S_SZ + srow] = c1[v] + be;
    }
}

// ---------------------------------------------------------------------------
// Kernel 2: per-row radix-select top-64, write dense z = relu(topk(pre))
// ---------------------------------------------------------------------------
__device__ __forceinline__ unsigned fkey(float v) {
    unsigned u = __float_as_uint(v);
    return (u & 0x80000000u) ? ~u : (u | 0x80000000u);   // monotonic key
}

__global__ __launch_bounds__(256) void sae_topk_kernel(
    const float* __restrict__ pre, float* __restrict__ z)
{
    const int b   = blockIdx.x;
    const int tid = threadIdx.x;
    __shared__ unsigned hist[256];
    __shared__ unsigned sh_pref;
    __shared__ int      sh_rem;
    __shared__ int      tie_take;

    const float* row = pre + (size_t)b * S_SZ;
    unsigned prefix = 0;
    int remaining = KTOP;

    for (int level = 3; level >= 0; --level) {
        hist[tid] = 0;
        __syncthreads();
        unsigned mask_hi = (level == 3) ? 0u : (0xFFFFFFFFu << ((level + 1) * 8));
        for (int s = tid; s < S_SZ; s += 256) {
            unsigned u = fkey(row[s]);
            if ((u & mask_hi) == (prefix & mask_hi))
                atomicAdd(&hist[(u >> (level * 8)) & 0xFF], 1u);
        }
        __syncthreads();
        if (tid == 0) {
            int rem = remaining;
            for (int bin = 255; bin >= 0; --bin) {
                int h = (int)hist[bin];
                if (rem <= h) {
                    sh_pref = prefix | ((unsigned)bin << (level * 8));
                    sh_rem  = rem;
                    break;
                }
                rem -= h;
            }
        }
        __syncthreads();
        prefix = sh_pref;
        remaining = sh_rem;
        __syncthreads();
    }

    if (tid == 0) tie_take = 0;
    __syncthreads();

    // selection pass: key > threshold always in; key == threshold takes first
    // `remaining` tickets (exact fp32 ties are measure-zero for this data).
    for (int s = tid; s < S_SZ; s += 256) {
        float v = row[s];
        unsigned u = fkey(v);
        bool take = u > prefix;
        if (u == prefix) {
            int t = atomicAdd(&tie_take, 1);
            take = (t < remaining);
        }
        z[(size_t)b * S_SZ + s] = take ? fmaxf(v, 0.0f) : 0.0f;
    }
}

// ---------------------------------------------------------------------------
// Kernel 3: decode + per-batch partial loss.
// Ordered (deterministic) compaction of positive z entries, then dense FMA.
// ---------------------------------------------------------------------------
__global__ __launch_bounds__(256) void sae_decode_kernel(
    const float* __restrict__ x, const float* __restrict__ W,
    const float* __restrict__ b_dec, const float* __restrict__ z,
    float* __restrict__ xhat, float* __restrict__ partial)
{
    const int b    = blockIdx.x;
    const int tid  = threadIdx.x;
    const int lane = tid & 31;
    const int wave = tid >> 5;

    __shared__ int   s_idx[KTOP];
    __shared__ float s_val[KTOP];
    __shared__ int   wcnt[8];
    __shared__ int   s_nsel;

    if (tid == 0) s_nsel = 0;
    __syncthreads();

    // ordered compaction of strictly-positive z entries (<= 64 of them)
    for (int chunk = 0; chunk < S_SZ; chunk += 256) {
        int s = chunk + tid;
        float zv = z[(size_t)b * S_SZ + s];
        bool take = zv > 0.0f;
        unsigned long long msk = __ballot(take);
        int wc = __popcll(msk);
        if (lane == 0) wcnt[wave] = wc;
        __syncthreads();
        int base = s_nsel;
        for (int w = 0; w < wave; ++w) base += wcnt[w];
        int before = __popcll(msk & ((1ull << lane) - 1ull));
        if (take) {
            int p = base + before;
            s_idx[p] = s;
            s_val[p] = zv;
        }
        __syncthreads();
        if (tid == 0) {
            int t = 0;
            for (int w = 0; w < 8; ++w) t += wcnt[w];
            s_nsel += t;
        }
        __syncthreads();
    }

    const int nsel = s_nsel;

    float acc[12];
    #pragma unroll
    for (int q = 0; q < 12; ++q) acc[q] = b_dec[q * 256 + tid];

    for (int j = 0; j < nsel; ++j) {
        const float* wr = W + (size_t)s_idx[j] * KDIM;
        float v = s_val[j];
        #pragma unroll
        for (int q = 0; q < 12; ++q)
            acc[q] = fmaf(v, wr[q * 256 + tid], acc[q]);
    }

    float sq = 0.0f;
    #pragma unroll
    for (int q = 0; q < 12; ++q) {
        int d = q * 256 + tid;
        float xv = x[(size_t)b * KDIM + d];
        float diff = acc[q] - xv;
        sq += diff * diff;
        xhat[(size_t)b * KDIM + d] = acc[q];
    }

    __shared__ float red[256];
    red[tid] = sq;
    __syncthreads();
    for (int off = 128; off > 0; off >>= 1) {
        if (tid < off) red[tid] += red[tid + off];
        __syncthreads();
    }
    if (tid == 0) partial[b] = red[0];
}

// ---------------------------------------------------------------------------
// Kernel 4: loss = sum(partial) / (B*T)
// ---------------------------------------------------------------------------
__global__ void sae_loss_kernel(const float* __restrict__ partial,
                                float* __restrict__ loss)
{
    if (threadIdx.x == 0) {
        float s = 0.0f;
        for (int b = 0; b < B_SZ; ++b) s += partial[b];
        loss[0] = s / (float)(B_SZ * T_SZ);
    }
}

// ---------------------------------------------------------------------------
extern "C" void kernel_launch(void* const* d_in, const int* in_sizes, int n_in,
                              void* d_out, int out_size, void* d_ws, size_t ws_size,
                              hipStream_t stream)
{
    const float* x     = (const float*)d_in[0];   // [32,4,768]
    const float* W     = (const float*)d_in[1];   // [65536,4,768]
    const float* b_enc = (const float*)d_in[2];   // [65536]
    const float* b_dec = (const float*)d_in[3];   // [4,768]

    float* out  = (float*)d_out;
    float* loss = out;                            // 1
    float* xhat = out + 1;                        // 32*3072
    float* z    = out + 1 + (size_t)B_SZ * KDIM;  // 32*65536

    float* pre     = (float*)d_ws;                       // 32*65536 floats (8 MB)
    float* partial = pre + (size_t)B_SZ * S_SZ;          // 32 floats

    sae_encode_kernel<<<S_SZ / 128, 256, 0, stream>>>(x, W, b_enc, pre);
    sae_topk_kernel<<<B_SZ, 256, 0, stream>>>(pre, z);
    sae_decode_kernel<<<B_SZ, 256, 0, stream>>>(x, W, b_dec, z, xhat, partial);
    sae_loss_kernel<<<1, 32, 0, stream>>>(partial, loss);
}